// MultiFrequencyConformer_28226525069925
// MI455X (gfx1250) — compile-verified
//
#include <hip/hip_runtime.h>
#include <hip/hip_bf16.h>
#include <math.h>

// ---------------- problem constants ----------------
#define BB   8
#define TT   256
#define MEL  128
#define DD   192
#define NC   20
#define NTHREADS 384            // 12 waves of 32
#define NWAVES   12
#define STATE_ELEMS (BB*DD*DD)  // 294912
#define PACK_TILE  3072         // floats per 16-col packed panel (48 groups * 64)

typedef __attribute__((ext_vector_type(2))) float v2f;
typedef __attribute__((ext_vector_type(8))) float v8f;
typedef __attribute__((ext_vector_type(4))) unsigned int u32x4;
typedef __attribute__((ext_vector_type(8))) int i32x8;
typedef __attribute__((ext_vector_type(4))) int i32x4;

// ---------------------------------------------------------------------------
// TDM: async 2D tile load  global -> LDS  (tile 8 rows x 192 f32, row stride
// T*D elements).  Descriptor per CDNA5 ISA 8.3/8.4; tracked by TENSORcnt.
// ---------------------------------------------------------------------------
__device__ inline void tdm_load_xt(const float* __restrict__ gsrc, float* lds_dst)
{
    const unsigned long long ga = (unsigned long long)(uintptr_t)gsrc;
    const unsigned int ldsa = (unsigned int)(uintptr_t)lds_dst;   // LDS byte offset

    u32x4 g0;
    g0[0] = 1u;                                            // count=1, user mode
    g0[1] = ldsa;                                          // lds_addr
    g0[2] = (unsigned int)(ga & 0xffffffffu);              // global_addr[31:0]
    g0[3] = (unsigned int)((ga >> 32) & 0x01ffffffu)       // global_addr[56:32]
            | 0x80000000u;                                 // type=2 ("image")

    i32x8 g1;
    g1[0] = 0x20000;                 // workgroup_mask=0, data_size=2 (4 bytes)
    g1[1] = 0;                       // tensor_dim0[15:0]=0 (dim0 = 0x60000)
    g1[2] = (int)((0x60000u >> 16)   // tensor_dim0[31:16] = 6
            | (8u << 16));           // tensor_dim1[15:0]  = 8
    g1[3] = (int)(192u << 16);       // tensor_dim1 hi=0 ; tile_dim0 = 192
    g1[4] = 8;                       // tile_dim1 = 8 ; tile_dim2 = 0
    g1[5] = (int)49152u;             // tensor_dim0_stride = T*D
    g1[6] = 0;
    g1[7] = 0;
    i32x4 z = {0, 0, 0, 0};          // groups 2/3 unused (<=2D tensor)
#if defined(__clang_major__) && (__clang_major__ >= 23)
    i32x8 z8 = {};
    __builtin_amdgcn_tensor_load_to_lds(g0, g1, z, z, z8, 0);
#else
    __builtin_amdgcn_tensor_load_to_lds(g0, g1, z, z, 0);
#endif
}

// ---------------------------------------------------------------------------
// Weight repack:  row-major [K=192][N]  ->  WMMA-native panels.
// packed idx = (((nt*48 + g)*2 + s)*16 + n)*2 + j   holds  W[4g+2s+j][nt*16+n]
// => per (lane, WMMA) the B fragment is ONE aligned b64; a wave's fetch per
//    WMMA is one contiguous 256B block.
// ---------------------------------------------------------------------------
__global__ void k_pack_w(const float* __restrict__ src, float* __restrict__ dst,
                         int N, int total)
{
    int idx = blockIdx.x * blockDim.x + threadIdx.x;
    if (idx >= total) return;
    const int j     = idx & 1;
    const int n     = (idx >> 1) & 15;
    const int s     = (idx >> 5) & 1;
    const int g     = (idx >> 6) % 48;
    const int nt    = idx / PACK_TILE;
    const int k     = 4 * g + 2 * s + j;
    dst[idx] = src[(size_t)k * N + nt * 16 + n];
}

// ---------------------------------------------------------------------------
// WMMA f32 16x16x4 tile from packed weights:  C[16,16] = A[16,192] * Wp(tile nt)
// ---------------------------------------------------------------------------
__device__ inline v8f wmma_tile_packed(const float* __restrict__ A, int lda,
                                       const float* __restrict__ Wp, int nt)
{
    const int lane = threadIdx.x & 31;
    const int m    = lane & 15;
    const int s    = lane >> 4;          // k-half select
    const float* wp = Wp + (size_t)nt * PACK_TILE + s * 32 + (m << 1);
    v8f c = {};
    for (int g = 0; g < 48; ++g) {       // K = 192, 4 per WMMA
        v2f a, b;
        a.x = A[m * lda + 4 * g + 2 * s];
        a.y = A[m * lda + 4 * g + 2 * s + 1];
        b   = *(const v2f*)wp;           // single b64, wave-contiguous 256B
        wp += 64;
        c = __builtin_amdgcn_wmma_f32_16x16x4_f32(false, a, false, b,
                                                  (short)0, c, false, false);
    }
    return c;
}

// ---------------------------------------------------------------------------
// xp = x @ W_in + b_in :  [2048,128] x [128,192] -> [2048,192]  (one-shot)
// ---------------------------------------------------------------------------
__global__ __launch_bounds__(32)
void k_xp_gemm(const float* __restrict__ x, const float* __restrict__ W,
               const float* __restrict__ bias, float* __restrict__ xp)
{
    const int tile = blockIdx.x;
    const int m0 = (tile / 12) * 16;
    const int n0 = (tile % 12) * 16;
    const int lane = threadIdx.x & 31;
    const int m  = lane & 15;
    const int kh = (lane >> 4) << 1;

    v8f c = {};
    for (int k = 0; k < MEL; k += 4) {
        v2f a, b;
        a.x = x[(size_t)(m0 + m) * MEL + k + kh];
        a.y = x[(size_t)(m0 + m) * MEL + k + kh + 1];
        const float* wp = W + (size_t)(k + kh) * DD + n0 + m;
        b.x = wp[0];
        b.y = wp[DD];
        c = __builtin_amdgcn_wmma_f32_16x16x4_f32(false, a, false, b,
                                                  (short)0, c, false, false);
    }
    const int col   = n0 + (lane & 15);
    const int rbase = (lane >> 4) * 8;
    const float bb  = bias[col];
#pragma unroll
    for (int r = 0; r < 8; ++r)
        xp[(size_t)(m0 + rbase + r) * DD + col] = c[r] + bb;
}

// ---------------------------------------------------------------------------
__global__ void k_zero(float* p, int n)
{
    int i = blockIdx.x * blockDim.x + threadIdx.x;
    if (i < n) p[i] = 0.0f;
}

// ---------------------------------------------------------------------------
struct ScanArgs {
    const float *xp;
    const float *pWv_a, *bv_a, *pWo_a, *bo_a;       // pW* = packed panels
    const float *ln1_g, *ln1_b;
    const float *pPw1, *pw1_b, *dw_w, *dw_b, *bn_s, *bn_b, *pPw2, *pw2_b;
    const float *pWk_t, *pWv_t, *pWq_t, *pWk_d, *pWv_d, *pWq_d;
    const float *ln2_g, *ln2_b, *Wc, *bc;
    float *Mt, *St, *Md, *Sd;
    float *out;
};

__device__ inline void wg_gemm(const float* __restrict__ A /*LDS [16][192]*/,
                               const float* __restrict__ Wp,
                               const float* __restrict__ bias,
                               float* __restrict__ dst /*LDS, rows 0..7*/,
                               int ldo, int N)
{
    const int wav  = threadIdx.x >> 5;
    const int lane = threadIdx.x & 31;
    for (int nt = wav; nt < (N >> 4); nt += NWAVES) {
        v8f c = wmma_tile_packed(A, DD, Wp, nt);
        if (lane < 16) {                 // lanes 0-15 hold rows 0..7 (real batches)
            const int col = nt * 16 + lane;
            const float bb = bias ? bias[col] : 0.0f;
#pragma unroll
            for (int r = 0; r < 8; ++r)
                dst[r * ldo + col] = c[r] + bb;
        }
    }
}

// dst[b,e] = (vsub ? vsub[b,e] - dot : dot),  dot = vec[b,:] . M[b,:,e]
// float2 along e: every M access is an aligned b64, wave-coalesced.
__device__ inline void wg_matvec(const float* __restrict__ vec, int ldv,
                                 const float* __restrict__ M,
                                 const float* __restrict__ vsub, int ldsub,
                                 float* __restrict__ dst)
{
    for (int pp = threadIdx.x; pp < (BB * DD) / 2; pp += NTHREADS) {
        const int b = pp / (DD / 2);
        const int e = (pp % (DD / 2)) * 2;
        const float* Mb = M + (size_t)b * DD * DD + e;
        const float* kb = vec + b * ldv;
        float a0 = 0.0f, a1 = 0.0f;
#pragma unroll 4
        for (int d = 0; d < DD; ++d) {
            const v2f mv = *(const v2f*)(Mb + (size_t)d * DD);
            const float kk = kb[d];
            a0 += kk * mv.x;
            a1 += kk * mv.y;
        }
        const int p = b * DD + e;
        if (vsub) {
            dst[p]     = vsub[b * ldsub + e]     - a0;
            dst[p + 1] = vsub[b * ldsub + e + 1] - a1;
        } else {
            dst[p]     = a0;
            dst[p + 1] = a1;
        }
    }
}

__device__ inline void wg_state_update(const float* __restrict__ kvec, int ldk,
                                       const float* __restrict__ err,
                                       float* __restrict__ S, float* __restrict__ M)
{
    for (int ii = threadIdx.x; ii < (BB * DD * DD) / 2; ii += NTHREADS) {
        const int idx = ii * 2;
        const int b = idx / (DD * DD);
        const int r = idx % (DD * DD);
        const int d = r / DD, e = r % DD;           // e even
        const float kk = 0.1f * kvec[b * ldk + d];
        v2f sv = *(v2f*)(S + idx);
        v2f mv = *(v2f*)(M + idx);
        const v2f ev = *(const v2f*)(err + b * DD + e);
        sv.x = 0.9f * sv.x + kk * ev.x;
        sv.y = 0.9f * sv.y + kk * ev.y;
        mv.x = 0.99f * mv.x + sv.x;
        mv.y = 0.99f * mv.y + sv.y;
        *(v2f*)(S + idx) = sv;
        *(v2f*)(M + idx) = mv;
    }
}

__device__ inline void wg_layernorm(const float* __restrict__ src /*[8][192]*/,
                                    const float* __restrict__ g,
                                    const float* __restrict__ bt,
                                    float* __restrict__ dst, bool accumulate)
{
    const int wav = threadIdx.x >> 5;
    const int lane = threadIdx.x & 31;
    if (wav < BB) {
        const float* row = src + wav * DD;
        float s = 0.0f;
        for (int i = lane; i < DD; i += 32) s += row[i];
        for (int o = 16; o > 0; o >>= 1) s += __shfl_xor(s, o, 32);
        const float mean = s * (1.0f / DD);
        float v = 0.0f;
        for (int i = lane; i < DD; i += 32) { float d = row[i] - mean; v += d * d; }
        for (int o = 16; o > 0; o >>= 1) v += __shfl_xor(v, o, 32);
        const float inv = rsqrtf(v * (1.0f / DD) + 1e-5f);
        for (int i = lane; i < DD; i += 32) {
            const float val = (row[i] - mean) * inv * g[i] + bt[i];
            if (accumulate) dst[wav * DD + i] += val;
            else            dst[wav * DD + i]  = val;
        }
    }
}

__global__ __launch_bounds__(NTHREADS, 1)
void k_scan(ScanArgs a)
{
    __shared__ float sA[16 * DD];        // residual stream / A operand (rows 8..15 == 0)
    __shared__ float sB[16 * DD];        // secondary A operand         (rows 8..15 == 0)
    __shared__ float sOut[8 * 576];      // GEMM outputs (kvq fused / pw1 / 192-wide)
    __shared__ float sErr[8 * DD];       // err, then reused for q@M retrieval
    __shared__ float sPool[8 * DD];
    __shared__ float sX[8 * DD];         // TDM staging tile for x_t

    // kick off the TDM prefetch of x_0 while we zero-init LDS
    if (threadIdx.x < 32) tdm_load_xt(a.xp, sX);

    for (int i = threadIdx.x; i < 16 * DD; i += NTHREADS) { sA[i] = 0.0f; sB[i] = 0.0f; }
    for (int i = threadIdx.x; i < 8 * DD;  i += NTHREADS) sPool[i] = 0.0f;
    __syncthreads();

    for (int t = 0; t < TT; ++t) {
        // --- consume the TDM tile for step t, prefetch t+1 ---
        if (threadIdx.x < 32) __builtin_amdgcn_s_wait_tensorcnt((short)0);
        __syncthreads();                                   // publish sX to all waves
        for (int p = threadIdx.x; p < BB * DD; p += NTHREADS) sA[p] = sX[p];
        __syncthreads();                                   // sX fully consumed
        if (threadIdx.x < 32) {
            if (t + 1 < TT) tdm_load_xt(a.xp + (size_t)(t + 1) * DD, sX);
            __builtin_prefetch(a.Md, 0, 1);                // global_prefetch_b8
            __builtin_prefetch(a.Mt, 0, 1);
        }

        // ---- delta (scene) memory: kvq, err, update (t%64==0), retrieve ----
        wg_gemm(sA, a.pWk_d, nullptr, sOut + 0,   576, DD);
        wg_gemm(sA, a.pWv_d, nullptr, sOut + 192, 576, DD);
        wg_gemm(sA, a.pWq_d, nullptr, sOut + 384, 576, DD);
        __syncthreads();
        wg_matvec(sOut + 0, 576, a.Md, sOut + 192, 576, sErr);   // err = v - k@Md
        __syncthreads();
        if ((t & 63) == 0) { wg_state_update(sOut + 0, 576, sErr, a.Sd, a.Md); __syncthreads(); }
        wg_matvec(sOut + 384, 576, a.Md, nullptr, 0, sErr);      // out = q@Md
        __syncthreads();
        for (int p = threadIdx.x; p < BB * DD; p += NTHREADS) sA[p] += 0.5f * sErr[p]; // x1
        __syncthreads();

        // ---- single-token attention: (x1 Wv_a + bv_a) Wo_a + bo_a ----
        wg_gemm(sA, a.pWv_a, a.bv_a, sOut, DD, DD);
        __syncthreads();
        for (int p = threadIdx.x; p < BB * DD; p += NTHREADS) sB[p] = sOut[p];
        __syncthreads();
        wg_gemm(sB, a.pWo_a, a.bo_a, sOut, DD, DD);
        __syncthreads();
        for (int p = threadIdx.x; p < BB * DD; p += NTHREADS) sA[p] += sOut[p];        // x2
        __syncthreads();

        // ---- conformer conv module ----
        wg_layernorm(sA, a.ln1_g, a.ln1_b, sB, false);
        __syncthreads();
        wg_gemm(sB, a.pPw1, a.pw1_b, sOut, 2 * DD, 2 * DD);
        __syncthreads();
        for (int p = threadIdx.x; p < BB * DD; p += NTHREADS) {
            const int b = p / DD, d = p % DD;
            const float aa = sOut[b * 384 + d];
            const float gg = sOut[b * 384 + DD + d];
            float h = aa * (1.0f / (1.0f + expf(-gg)));            // GLU
            h = h * a.dw_w[d] + a.dw_b[d];                         // depthwise center tap
            h = h * a.bn_s[d] + a.bn_b[d];                         // eval BN
            h = h * (1.0f / (1.0f + expf(-h)));                    // SiLU
            sB[p] = h;
        }
        __syncthreads();
        wg_gemm(sB, a.pPw2, a.pw2_b, sOut, DD, DD);
        __syncthreads();
        for (int p = threadIdx.x; p < BB * DD; p += NTHREADS) sA[p] += sOut[p];        // x3
        __syncthreads();

        // ---- theta (event) memory ----
        wg_gemm(sA, a.pWk_t, nullptr, sOut + 0,   576, DD);
        wg_gemm(sA, a.pWv_t, nullptr, sOut + 192, 576, DD);
        wg_gemm(sA, a.pWq_t, nullptr, sOut + 384, 576, DD);
        __syncthreads();
        wg_matvec(sOut + 0, 576, a.Mt, sOut + 192, 576, sErr);
        __syncthreads();
        if ((t & 7) == 0) { wg_state_update(sOut + 0, 576, sErr, a.St, a.Mt); __syncthreads(); }
        wg_matvec(sOut + 384, 576, a.Mt, nullptr, 0, sErr);
        __syncthreads();
        for (int p = threadIdx.x; p < BB * DD; p += NTHREADS) sA[p] += 0.5f * sErr[p]; // x4
        __syncthreads();

        wg_layernorm(sA, a.ln2_g, a.ln2_b, sPool, true);           // pooled += ln2(x4)
        __syncthreads();
    }

    // ---- classifier: (pooled/T) @ Wc + bc -> [8,20] ----
    if (threadIdx.x < BB * NC) {
        const int b = threadIdx.x / NC, c = threadIdx.x % NC;
        float acc = a.bc[c];
        for (int d = 0; d < DD; ++d)
            acc += sPool[b * DD + d] * (1.0f / TT) * a.Wc[d * NC + c];
        a.out[b * NC + c] = acc;
    }
}

// ---------------------------------------------------------------------------
extern "C" void kernel_launch(void* const* d_in, const int* in_sizes, int n_in,
                              void* d_out, int out_size, void* d_ws, size_t ws_size,
                              hipStream_t stream)
{
    const float* x     = (const float*)d_in[0];
    const float* W_in  = (const float*)d_in[1];
    const float* b_in  = (const float*)d_in[2];

    float* ws = (float*)d_ws;
    float* xp = ws;                                    // [B*T, D]
    float* Mt = ws + (size_t)BB * TT * DD;
    float* St = Mt + STATE_ELEMS;
    float* Md = St + STATE_ELEMS;
    float* Sd = Md + STATE_ELEMS;
    float* pk = Sd + STATE_ELEMS;                      // packed weight arena

    const int SQ = DD * DD;                            // 36864 (packed sz, N=192)
    float* pWv_a = pk;           float* pWo_a = pWv_a + SQ;
    float* pPw1  = pWo_a + SQ;   /* 192x384 -> 73728 */
    float* pPw2  = pPw1 + 2*SQ;
    float* pWk_t = pPw2 + SQ;    float* pWv_t = pWk_t + SQ;  float* pWq_t = pWv_t + SQ;
    float* pWk_d = pWq_t + SQ;   float* pWv_d = pWk_d + SQ;  float* pWq_d = pWv_d + SQ;

    // zero the four state matrices
    {
        const int n = 4 * STATE_ELEMS;
        k_zero<<<(n + 255) / 256, 256, 0, stream>>>(Mt, n);
    }

    // one-shot weight repacking into WMMA-native panels (amortized over 256 steps)
    {
        const int g1 = (SQ + 255) / 256, g2 = (2 * SQ + 255) / 256;
        k_pack_w<<<g1, 256, 0, stream>>>((const float*)d_in[3],  pWv_a, DD,     SQ);
        k_pack_w<<<g1, 256, 0, stream>>>((const float*)d_in[5],  pWo_a, DD,     SQ);
        k_pack_w<<<g2, 256, 0, stream>>>((const float*)d_in[9],  pPw1,  2 * DD, 2 * SQ);
        k_pack_w<<<g1, 256, 0, stream>>>((const float*)d_in[15], pPw2,  DD,     SQ);
        k_pack_w<<<g1, 256, 0, stream>>>((const float*)d_in[17], pWk_t, DD,     SQ);
        k_pack_w<<<g1, 256, 0, stream>>>((const float*)d_in[18], pWv_t, DD,     SQ);
        k_pack_w<<<g1, 256, 0, stream>>>((const float*)d_in[19], pWq_t, DD,     SQ);
        k_pack_w<<<g1, 256, 0, stream>>>((const float*)d_in[20], pWk_d, DD,     SQ);
        k_pack_w<<<g1, 256, 0, stream>>>((const float*)d_in[21], pWv_d, DD,     SQ);
        k_pack_w<<<g1, 256, 0, stream>>>((const float*)d_in[22], pWq_d, DD,     SQ);
    }

    // xp = x @ W_in + b_in  (WMMA f32 GEMM, one wave per tile)
    k_xp_gemm<<<(BB * TT / 16) * (DD / 16), 32, 0, stream>>>(x, W_in, b_in, xp);

    ScanArgs sa;
    sa.xp    = xp;
    sa.pWv_a = pWv_a;                 sa.bv_a = (const float*)d_in[4];
    sa.pWo_a = pWo_a;                 sa.bo_a = (const float*)d_in[6];
    sa.ln1_g = (const float*)d_in[7]; sa.ln1_b= (const float*)d_in[8];
    sa.pPw1  = pPw1;                  sa.pw1_b= (const float*)d_in[10];
    sa.dw_w  = (const float*)d_in[11]; sa.dw_b = (const float*)d_in[12];
    sa.bn_s  = (const float*)d_in[13]; sa.bn_b = (const float*)d_in[14];
    sa.pPw2  = pPw2;                  sa.pw2_b= (const float*)d_in[16];
    sa.pWk_t = pWk_t; sa.pWv_t = pWv_t; sa.pWq_t = pWq_t;
    sa.pWk_d = pWk_d; sa.pWv_d = pWv_d; sa.pWq_d = pWq_d;
    sa.ln2_g = (const float*)d_in[23]; sa.ln2_b= (const float*)d_in[24];
    sa.Wc    = (const float*)d_in[25]; sa.bc   = (const float*)d_in[26];
    sa.Mt = Mt; sa.St = St; sa.Md = Md; sa.Sd = Sd;
    sa.out = (float*)d_out;

    k_scan<<<1, NTHREADS, 0, stream>>>(sa);
}